// MCILayer_83279415869730
// MI455X (gfx1250) — compile-verified
//
#include <hip/hip_runtime.h>
#include <hip/hip_bf16.h>

typedef __attribute__((ext_vector_type(2))) float v2f;
typedef __attribute__((ext_vector_type(8))) float v8f;

#define DIMX 768
#define NTOK 32768   // 8 * 4096 concatenated tokens
#define DI 16
#define DS 16

static __device__ inline v8f wmma_f32_4(v2f a, v2f b, v8f c) {
  // D = A(16x4 f32) * B(4x16 f32) + C(16x16 f32), full fp32 matrix op
  return __builtin_amdgcn_wmma_f32_16x16x4_f32(
      false, a, false, b, (short)0, c, false, false);
}

// ---------------------------------------------------------------------------
// Down-projection: cat[t,0:8] = src[t,:]@W + b   (K = 768, fp32 WMMA 16x16x4)
// Block = 256 thr = 8 waves, handles 128 consecutive cat-tokens (one 16-token
// tile per wave).  W (768x8, padded to 16 cols) staged in LDS as K-pairs.
// ---------------------------------------------------------------------------
__global__ __launch_bounds__(256) void down_proj_kernel(
    const float* __restrict__ x, const float* __restrict__ xi,
    const float* __restrict__ W_dx, const float* __restrict__ b_dx,
    const float* __restrict__ W_dxi, const float* __restrict__ b_dxi,
    float* __restrict__ cat) {
  __shared__ v2f wlds[384 * 16];  // pair[(k>>1)][n] = {W[k,n], W[k+1,n]}, 48KB
  int tok0 = blockIdx.x * 128;
  int r = tok0 & 4095;
  bool isX = (r < 2048);
  const float* W = isX ? W_dx : W_dxi;
  const float* bias = isX ? b_dx : b_dxi;
  const float* src = isX ? x : xi;
  int b = tok0 >> 12;
  int srow0 = b * 2048 + (isX ? r : (r - 2048));

  for (int i = threadIdx.x; i < 384 * 16; i += 256) {
    int n = i & 15, kk = i >> 4;
    v2f w;
    w.x = (n < 8) ? W[(2 * kk) * 8 + n] : 0.f;
    w.y = (n < 8) ? W[(2 * kk + 1) * 8 + n] : 0.f;
    wlds[i] = w;
  }
  __syncthreads();

  int wave = threadIdx.x >> 5;
  int lane = threadIdx.x & 31;
  int m = lane & 15;       // N for B/C; M row for A
  int o = lane >> 4;       // K half-select
  int tokTile = tok0 + wave * 16;
  const float* arow = src + (size_t)(srow0 + wave * 16 + m) * DIMX;

  v8f c = {};
  for (int k0 = 0; k0 < DIMX; k0 += 4) {
    v2f a = *(const v2f*)(arow + k0 + 2 * o);            // A: K=k0+2o, +1
    v2f bf = wlds[((k0 >> 1) + o) * 16 + m];             // B: rows 2o,2o+1
    c = wmma_f32_4(a, bf, c);
  }
  if (m < 8) {
    float bb = bias[m];
    #pragma unroll
    for (int rr = 0; rr < 8; ++rr) {
      int trow = tokTile + rr + 8 * o;                   // C: M = rr + 8*o
      cat[trow * 8 + m] = c[rr] + bb;
    }
  }
}

// ---------------------------------------------------------------------------
// in-projection: xz = cat @ W_in + b_in ; split into h_pre / z
// ---------------------------------------------------------------------------
__global__ void inproj_kernel(const float* __restrict__ cat,
                              const float* __restrict__ W_in,
                              const float* __restrict__ b_in,
                              float* __restrict__ h_pre, float* __restrict__ z) {
  int idx = blockIdx.x * blockDim.x + threadIdx.x;
  if (idx >= NTOK * 32) return;
  int t = idx >> 5, j = idx & 31;
  const float* u = cat + t * 8;
  float acc = b_in[j];
  #pragma unroll
  for (int k = 0; k < 8; ++k) acc += u[k] * W_in[k * 32 + j];
  if (j < DI) h_pre[t * DI + j] = acc;
  else        z[t * DI + (j - DI)] = acc;
}

// ---------------------------------------------------------------------------
// causal depthwise conv (K=4) + SiLU, then x-projection (W_xp) and dt
// ---------------------------------------------------------------------------
__global__ void conv_proj_kernel(const float* __restrict__ h_pre,
                                 const float* __restrict__ conv_w,
                                 const float* __restrict__ conv_b,
                                 const float* __restrict__ W_xp,
                                 const float* __restrict__ W_dt,
                                 const float* __restrict__ b_dt,
                                 float* __restrict__ h, float* __restrict__ dt,
                                 float* __restrict__ Bv, float* __restrict__ Cv) {
  int t = blockIdx.x * blockDim.x + threadIdx.x;
  if (t >= NTOK) return;
  int l = t & 4095;
  float hv[DI];
  #pragma unroll
  for (int d = 0; d < DI; ++d) {
    float acc = conv_b[d];
    #pragma unroll
    for (int k = 0; k < 4; ++k) {
      int ll = l - 3 + k;
      if (ll >= 0) acc += conv_w[d * 4 + k] * h_pre[(t - 3 + k) * DI + d];
    }
    float s = acc / (1.f + __expf(-acc));  // SiLU
    hv[d] = s;
    h[t * DI + d] = s;
  }
  float dtr = 0.f;
  #pragma unroll
  for (int d = 0; d < DI; ++d) dtr += hv[d] * W_xp[d * 33 + 0];
  #pragma unroll
  for (int s = 0; s < DS; ++s) {
    float accB = 0.f, accC = 0.f;
    #pragma unroll
    for (int d = 0; d < DI; ++d) {
      accB += hv[d] * W_xp[d * 33 + 1 + s];
      accC += hv[d] * W_xp[d * 33 + 17 + s];
    }
    Bv[t * DS + s] = accB;
    Cv[t * DS + s] = accC;
  }
  #pragma unroll
  for (int d = 0; d < DI; ++d) {
    float v = dtr * W_dt[d] + b_dt[d];
    dt[t * DI + d] = (v > 20.f) ? v : log1pf(__expf(v));  // softplus
  }
}

// ---------------------------------------------------------------------------
// selective scan: 1 block per batch, thread = (d,s) state element.
// Serial chain is a single FMA per step; y via 16-lane shfl_xor reduction.
// ---------------------------------------------------------------------------
__global__ __launch_bounds__(256) void scan_kernel(
    const float* __restrict__ h, const float* __restrict__ dt,
    const float* __restrict__ Bv, const float* __restrict__ Cv,
    const float* __restrict__ A_log, const float* __restrict__ Dp,
    const float* __restrict__ z, float* __restrict__ y_act) {
  __shared__ float h_c[64 * DI], dt_c[64 * DI], B_c[64 * DS], C_c[64 * DS];
  int b = blockIdx.x;
  int tid = threadIdx.x;
  int d = tid >> 4, s = tid & 15;          // lane = (d&1)*16 + s  -> s = lane%16
  float A = -__expf(A_log[d * DS + s]);
  float Dpd = Dp[d];
  float state = 0.f;
  size_t base = (size_t)b * 4096 * DI;

  for (int c0 = 0; c0 < 4096; c0 += 64) {
    __syncthreads();
    for (int i = tid; i < 64 * DI; i += 256) {
      size_t g = base + (size_t)c0 * DI + i;
      h_c[i] = h[g]; dt_c[i] = dt[g]; B_c[i] = Bv[g]; C_c[i] = Cv[g];
    }
    __syncthreads();
    for (int tt = 0; tt < 64; ++tt) {
      float hv = h_c[tt * DI + d];
      float dtv = dt_c[tt * DI + d];
      float dA = __expf(dtv * A);
      state = state * dA + (dtv * hv) * B_c[tt * DS + s];
      float contrib = state * C_c[tt * DS + s];
      contrib += __shfl_xor(contrib, 1, 32);
      contrib += __shfl_xor(contrib, 2, 32);
      contrib += __shfl_xor(contrib, 4, 32);
      contrib += __shfl_xor(contrib, 8, 32);
      if (s == 0) {
        int t = b * 4096 + c0 + tt;
        float y = contrib + Dpd * hv;
        float zv = z[(size_t)t * DI + d];
        y_act[(size_t)t * DI + d] = y * (zv / (1.f + __expf(-zv)));
      }
    }
  }
}

// ---------------------------------------------------------------------------
// out-projection: a_out = y_act @ W_out + b_out
// ---------------------------------------------------------------------------
__global__ void outproj_kernel(const float* __restrict__ y_act,
                               const float* __restrict__ W_out,
                               const float* __restrict__ b_out,
                               float* __restrict__ a_out) {
  int idx = blockIdx.x * blockDim.x + threadIdx.x;
  if (idx >= NTOK * 8) return;
  int t = idx >> 3, j = idx & 7;
  float acc = b_out[j];
  #pragma unroll
  for (int d = 0; d < DI; ++d) acc += y_act[t * DI + d] * W_out[d * 8 + j];
  a_out[idx] = acc;
}

// ---------------------------------------------------------------------------
// Up-projection + residual: out = src + a_out @ W + b  (K=8 -> 2 WMMAs/tile)
// Block = 1 token-tile (16 tokens), 8 waves sweep 48 col-tiles.
// ---------------------------------------------------------------------------
__global__ __launch_bounds__(256) void up_proj_kernel(
    const float* __restrict__ x, const float* __restrict__ xi,
    const float* __restrict__ a_out,
    const float* __restrict__ W_ix, const float* __restrict__ b_ix,
    const float* __restrict__ W_ixi, const float* __restrict__ b_ixi,
    float* __restrict__ out) {
  __shared__ v2f wlds[4 * DIMX];   // pair[k>>1][col] = {W[k,col], W[k+1,col]}
  __shared__ float blds[DIMX];
  int tok0 = blockIdx.x * 16;
  int r = tok0 & 4095;
  bool isX = (r < 2048);
  const float* W = isX ? W_ix : W_ixi;
  const float* bias = isX ? b_ix : b_ixi;
  const float* res = isX ? x : xi;
  int b = tok0 >> 12;
  int orow0 = b * 2048 + (isX ? r : (r - 2048));
  float* dst = out + (isX ? (size_t)0 : (size_t)16384 * DIMX);

  for (int i = threadIdx.x; i < 4 * DIMX; i += 256) {
    int col = i % DIMX, kk = i / DIMX;
    v2f w;
    w.x = W[(2 * kk) * DIMX + col];
    w.y = W[(2 * kk + 1) * DIMX + col];
    wlds[i] = w;
  }
  for (int i = threadIdx.x; i < DIMX; i += 256) blds[i] = bias[i];
  __syncthreads();

  int wave = threadIdx.x >> 5, lane = threadIdx.x & 31;
  int m = lane & 15, o = lane >> 4;
  const float* ar = a_out + (size_t)(tok0 + m) * 8;
  v2f a0 = *(const v2f*)(ar + 2 * o);          // K = 0..3 fragment
  v2f a1 = *(const v2f*)(ar + 4 + 2 * o);      // K = 4..7 fragment

  for (int ct = wave; ct < 48; ct += 8) {
    int n0 = ct * 16;
    v2f b0 = wlds[o * DIMX + n0 + m];          // rows 2o,2o+1 of K=0..3
    v2f b1 = wlds[(2 + o) * DIMX + n0 + m];    // rows 2o,2o+1 of K=4..7
    v8f c = {};
    c = wmma_f32_4(a0, b0, c);
    c = wmma_f32_4(a1, b1, c);
    float bb = blds[n0 + m];
    #pragma unroll
    for (int rr = 0; rr < 8; ++rr) {
      int row = orow0 + rr + 8 * o;
      size_t idx = (size_t)row * DIMX + n0 + m;
      dst[idx] = res[idx] + c[rr] + bb;
    }
  }
}

// ---------------------------------------------------------------------------
extern "C" void kernel_launch(void* const* d_in, const int* in_sizes, int n_in,
                              void* d_out, int out_size, void* d_ws, size_t ws_size,
                              hipStream_t stream) {
  (void)in_sizes; (void)n_in; (void)out_size; (void)ws_size;
  const float* x     = (const float*)d_in[0];
  const float* xi    = (const float*)d_in[1];
  const float* W_dx  = (const float*)d_in[2];
  const float* b_dx  = (const float*)d_in[3];
  const float* W_dxi = (const float*)d_in[4];
  const float* b_dxi = (const float*)d_in[5];
  const float* W_in  = (const float*)d_in[6];
  const float* b_in  = (const float*)d_in[7];
  const float* conv_w= (const float*)d_in[8];
  const float* conv_b= (const float*)d_in[9];
  const float* W_xp  = (const float*)d_in[10];
  const float* W_dt  = (const float*)d_in[11];
  const float* b_dt  = (const float*)d_in[12];
  const float* A_log = (const float*)d_in[13];
  const float* Dp    = (const float*)d_in[14];
  const float* W_out = (const float*)d_in[15];
  const float* b_out = (const float*)d_in[16];
  const float* W_ix  = (const float*)d_in[17];
  const float* b_ix  = (const float*)d_in[18];
  const float* W_ixi = (const float*)d_in[19];
  const float* b_ixi = (const float*)d_in[20];
  float* out = (float*)d_out;

  // workspace layout (floats): 16 MiB total
  float* ws    = (float*)d_ws;
  float* cat   = ws;               // 32768*8
  float* h_pre = cat   + 262144;   // 32768*16
  float* z     = h_pre + 524288;
  float* h     = z     + 524288;
  float* dtb   = h     + 524288;
  float* Bv    = dtb   + 524288;
  float* Cv    = Bv    + 524288;
  float* y_act = Cv    + 524288;
  float* a_out = y_act + 524288;   // 32768*8

  down_proj_kernel<<<256, 256, 0, stream>>>(x, xi, W_dx, b_dx, W_dxi, b_dxi, cat);
  inproj_kernel<<<4096, 256, 0, stream>>>(cat, W_in, b_in, h_pre, z);
  conv_proj_kernel<<<128, 256, 0, stream>>>(h_pre, conv_w, conv_b, W_xp, W_dt, b_dt,
                                            h, dtb, Bv, Cv);
  scan_kernel<<<8, 256, 0, stream>>>(h, dtb, Bv, Cv, A_log, Dp, z, y_act);
  outproj_kernel<<<1024, 256, 0, stream>>>(y_act, W_out, b_out, a_out);
  up_proj_kernel<<<2048, 256, 0, stream>>>(x, xi, a_out, W_ix, b_ix, W_ixi, b_ixi, out);
}